// GNNWithSoftTree_20169166422307
// MI455X (gfx1250) — compile-verified
//
#include <hip/hip_runtime.h>
#include <cstddef>

typedef float v2f __attribute__((ext_vector_type(2)));
typedef float v8f __attribute__((ext_vector_type(8)));

// Full-precision f32 WMMA: D = A(16x4) * B(4x16) + C(16x16)
#define WMMA_F32(a, b, c) \
  __builtin_amdgcn_wmma_f32_16x16x4_f32(false, (a), false, (b), (short)0, (c), false, false)

// ---------------------------------------------------------------- utilities
__global__ void k_zero(float* __restrict__ p, int n) {
  int i = blockIdx.x * blockDim.x + threadIdx.x;
  if (i < n) p[i] = 0.0f;
}

__global__ void k_degree(const int* __restrict__ dst, float* __restrict__ deg, int E) {
  int e = blockIdx.x * blockDim.x + threadIdx.x;
  if (e < E)
    __hip_atomic_fetch_add(&deg[dst[e]], 1.0f, __ATOMIC_RELAXED, __HIP_MEMORY_SCOPE_AGENT);
}

__global__ void k_dinv(float* __restrict__ dinv, int n) {
  int i = blockIdx.x * blockDim.x + threadIdx.x;
  if (i < n) dinv[i] = rsqrtf(dinv[i] + 1.0f);   // +1 self-loop => deg >= 1 always
}

// ---------------------------------------------------------------- WMMA GEMM
// C[n, NCOLS] = A[n, K] @ W[NCOLS, K]^T     (W row-major [out, in], PyG style)
// One wave owns a 16-row tile and accumulates all NCOLS/16 column tiles so A
// is streamed exactly once per row tile.
template <int K, int NCOLS>
__global__ void k_gemm_wmma(const float* __restrict__ A, const float* __restrict__ W,
                            float* __restrict__ C, int n) {
  const int lane = threadIdx.x & 31;
  const int wave = threadIdx.x >> 5;
  const int wavesPerBlock = blockDim.x >> 5;
  const int ntiles = (n + 15) >> 4;
  int tile = blockIdx.x * wavesPerBlock + wave;
  if (tile >= ntiles) tile = ntiles - 1;        // clamp: duplicate work, idempotent stores
  const int rowbase = tile << 4;

  const int m = lane & 15;                       // A row / B col within tile
  const int koff = (lane >> 4) << 1;             // lanes 16..31 hold K+2,K+3
  int arow = rowbase + m; if (arow >= n) arow = n - 1;

  constexpr int CT = NCOLS / 16;
  v8f acc[CT];
  #pragma unroll
  for (int t = 0; t < CT; ++t) acc[t] = {};

  for (int k = 0; k < K; k += 4) {
    v2f a = *(const v2f*)(A + (size_t)arow * K + k + koff);
    #pragma unroll
    for (int t = 0; t < CT; ++t) {
      v2f b = *(const v2f*)(W + (size_t)(t * 16 + m) * K + k + koff);  // B[k][n] = W[n][k]
      acc[t] = WMMA_F32(a, b, acc[t]);
    }
  }

  const int rrow = rowbase + ((lane >> 4) << 3); // lanes>=16 hold rows 8..15
  #pragma unroll
  for (int t = 0; t < CT; ++t)
    #pragma unroll
    for (int r = 0; r < 8; ++r) {
      int row = rrow + r;
      if (row < n) C[(size_t)row * NCOLS + t * 16 + m] = acc[t][r];
    }
}

// ---------------------------------------------------------------- aggregation
// acc[i,f] = dinv[i]^2 * h[i,f]   (self-loop contribution doubles as the init)
template <int F>
__global__ void k_selfloop(const float* __restrict__ h, const float* __restrict__ dinv,
                           float* __restrict__ acc, int n) {
  int i = blockIdx.x * blockDim.x + threadIdx.x;
  int node = i / F;
  if (node < n) {
    float d = dinv[node];
    acc[i] = d * d * h[i];
  }
}

// One edge per 32 lanes; PER features per lane (F == 32*PER), coalesced gathers.
template <int F, int PER>
__global__ void k_scatter(const int* __restrict__ src, const int* __restrict__ dst,
                          const float* __restrict__ dinv, const float* __restrict__ h,
                          float* __restrict__ acc, int E) {
  int gid = blockIdx.x * blockDim.x + threadIdx.x;
  int e = gid >> 5;
  int lane = gid & 31;
  if (e >= E) return;
  int s = src[e], d = dst[e];
  float w = dinv[s] * dinv[d];
  #pragma unroll
  for (int p = 0; p < PER; ++p) {
    int f = lane + p * 32;
    float v = w * h[(size_t)s * F + f];
    __hip_atomic_fetch_add(&acc[(size_t)d * F + f], v,
                           __ATOMIC_RELAXED, __HIP_MEMORY_SCOPE_AGENT);
  }
}

__global__ void k_bias_relu64(const float* __restrict__ acc, const float* __restrict__ b,
                              float* __restrict__ out, int total) {
  int i = blockIdx.x * blockDim.x + threadIdx.x;
  if (i < total) {
    float v = acc[i] + b[i & 63];
    out[i] = v > 0.0f ? v : 0.0f;
  }
}

// ---------------------------------------------------------------- fused tail
// gates = sigmoid((a2 + b2) @ gate_w^T + gate_b); out = gates @ leaf_w
// WMMA for gates (K=32, two 16-col tiles), LDS re-layout, scalar 32x8 leaf GEMM.
__global__ void k_tail(const float* __restrict__ a2, const float* __restrict__ b2,
                       const float* __restrict__ gate_w, const float* __restrict__ gate_b,
                       const float* __restrict__ leaf_w, float* __restrict__ out, int n) {
  __shared__ float lds[4][16][33];               // +1 pad: no bank conflicts
  const int lane = threadIdx.x & 31;
  const int wave = threadIdx.x >> 5;
  const int ntiles = (n + 15) >> 4;
  int tile = blockIdx.x * 4 + wave;
  if (tile >= ntiles) tile = ntiles - 1;
  const int rowbase = tile << 4;

  const int m = lane & 15;
  const int koff = (lane >> 4) << 1;
  int arow = rowbase + m; if (arow >= n) arow = n - 1;

  v8f c0 = {}, c1 = {};
  for (int k = 0; k < 32; k += 4) {
    v2f a;
    a.x = a2[(size_t)arow * 32 + k + koff]     + b2[k + koff];
    a.y = a2[(size_t)arow * 32 + k + koff + 1] + b2[k + koff + 1];
    v2f bA = *(const v2f*)(gate_w + (size_t)m * 32 + k + koff);          // cols 0..15
    v2f bB = *(const v2f*)(gate_w + (size_t)(16 + m) * 32 + k + koff);   // cols 16..31
    c0 = WMMA_F32(a, bA, c0);
    c1 = WMMA_F32(a, bB, c1);
  }

  const int rrow = (lane >> 4) << 3;
  const float gb0 = gate_b[m], gb1 = gate_b[16 + m];
  #pragma unroll
  for (int r = 0; r < 8; ++r) {
    float g0 = 1.0f / (1.0f + __expf(-(c0[r] + gb0)));
    float g1 = 1.0f / (1.0f + __expf(-(c1[r] + gb1)));
    lds[wave][rrow + r][m]      = g0;
    lds[wave][rrow + r][16 + m] = g1;
  }
  __syncthreads();

  if (lane < 16) {
    int row = rowbase + lane;
    if (row < n) {
      float o[8] = {0, 0, 0, 0, 0, 0, 0, 0};
      #pragma unroll
      for (int k2 = 0; k2 < 32; ++k2) {
        float g = lds[wave][lane][k2];
        #pragma unroll
        for (int c = 0; c < 8; ++c) o[c] += g * leaf_w[k2 * 8 + c];
      }
      #pragma unroll
      for (int c = 0; c < 8; ++c) out[(size_t)row * 8 + c] = o[c];
    }
  }
}

// ---------------------------------------------------------------- launcher
extern "C" void kernel_launch(void* const* d_in, const int* in_sizes, int n_in,
                              void* d_out, int out_size, void* d_ws, size_t ws_size,
                              hipStream_t stream) {
  const float* x      = (const float*)d_in[0];   // [n,128]
  const int*   edge   = (const int*)d_in[1];     // [2,E]
  const float* W1     = (const float*)d_in[2];   // [64,128]
  const float* b1     = (const float*)d_in[3];   // [64]
  const float* W2     = (const float*)d_in[4];   // [32,64]
  const float* b2     = (const float*)d_in[5];   // [32]
  const float* gate_w = (const float*)d_in[6];   // [32,32]
  const float* gate_b = (const float*)d_in[7];   // [32]
  const float* leaf_w = (const float*)d_in[8];   // [32,8]

  const int n = in_sizes[0] / 128;
  const int E = in_sizes[1] / 2;
  const int* src = edge;
  const int* dst = edge + E;

  float* ws   = (float*)d_ws;
  float* dinv = ws;                               // n
  float* h1   = dinv + (size_t)n;                 // n*64
  float* a1   = h1 + (size_t)n * 64;              // n*64
  float* t2   = a1 + (size_t)n * 64;              // n*32
  float* a2   = t2 + (size_t)n * 32;              // n*32

  const int ntiles = (n + 15) / 16;

  // degree -> dinv
  k_zero<<<(n + 255) / 256, 256, 0, stream>>>(dinv, n);
  k_degree<<<(E + 255) / 256, 256, 0, stream>>>(dst, dinv, E);
  k_dinv<<<(n + 255) / 256, 256, 0, stream>>>(dinv, n);

  // layer 1: GEMM -> self-loop init -> scatter -> bias+relu
  k_gemm_wmma<128, 64><<<(ntiles + 7) / 8, 256, 0, stream>>>(x, W1, h1, n);
  k_selfloop<64><<<((size_t)n * 64 + 255) / 256, 256, 0, stream>>>(h1, dinv, a1, n);
  k_scatter<64, 2><<<(unsigned)(((size_t)E * 32 + 255) / 256), 256, 0, stream>>>(
      src, dst, dinv, h1, a1, E);
  k_bias_relu64<<<((size_t)n * 64 + 255) / 256, 256, 0, stream>>>(a1, b1, h1, n * 64);

  // layer 2: GEMM -> self-loop init -> scatter  (bias b2 folded into tail)
  k_gemm_wmma<64, 32><<<(ntiles + 7) / 8, 256, 0, stream>>>(h1, W2, t2, n);
  k_selfloop<32><<<((size_t)n * 32 + 255) / 256, 256, 0, stream>>>(t2, dinv, a2, n);
  k_scatter<32, 1><<<(unsigned)(((size_t)E * 32 + 255) / 256), 256, 0, stream>>>(
      src, dst, dinv, t2, a2, E);

  // fused soft-tree head
  k_tail<<<(ntiles + 3) / 4, 128, 0, stream>>>(a2, b2, gate_w, gate_b, leaf_w,
                                               (float*)d_out, n);
}